// OnehotGather_5952824672601
// MI455X (gfx1250) — compile-verified
//
#include <hip/hip_runtime.h>

// ---------------------------------------------------------------------------
// OnehotGather: out[b,p,h] = sequence[b, positions[b,p], h]
//   B=8, S=4096, H=1024, P=512  (float32 sequence/out, int64 positions)
// Pure bandwidth-bound row gather (~32 MB traffic -> ~1.4us @ 23.3 TB/s).
// Implemented with the CDNA5 Tensor Data Mover in GATHER MODE:
//   tensor_load_to_lds  (gather_mode=1, 8 x 32-bit row indices, 4KB rows)
//   s_wait_tensorcnt 0
//   tensor_store_from_lds (contiguous 32KB tile -> out)
// ---------------------------------------------------------------------------

typedef unsigned int u32;
typedef unsigned long long u64;
typedef u32 v4u __attribute__((ext_vector_type(4)));
typedef int v8i __attribute__((ext_vector_type(8)));
typedef int v4i __attribute__((ext_vector_type(4)));

static constexpr int kB = 8;
static constexpr int kS = 4096;
static constexpr int kH = 1024;   // row length (floats)
static constexpr int kP = 512;
static constexpr int kRows = 8;   // rows per TDM gather (32-bit index mode max)
static constexpr int kTileElems = kRows * kH;  // 8192 floats = 32 KB

__global__ __launch_bounds__(32)
void OnehotGather_tdm_kernel(const float* __restrict__ seq,
                             const long long* __restrict__ pos,
                             float* __restrict__ out) {
    __shared__ float tile[kTileElems];            // 32 KB staging in LDS

    const u32 tileId = blockIdx.x;                // 0 .. B*P/kRows - 1
    const u32 b      = tileId / (kP / kRows);
    const u32 p0     = (tileId % (kP / kRows)) * kRows;

    // Low 32 bits of the int64 positions (values < S=4096, so lo word is it).
    const u32* pos32 = (const u32*)pos;
    int idx[kRows];
#pragma unroll
    for (int i = 0; i < kRows; ++i)
        idx[i] = (int)pos32[2u * (b * kP + p0 + (u32)i)];

    const u32 ldsAddr = (u32)(unsigned long long)(uintptr_t)&tile[0];

    // ---------------- Gather load descriptor: seq[b, idx[0..7], :] -> LDS --
    const u64 gsrc = (u64)(uintptr_t)(seq + (u64)b * (u64)(kS * kH));
    v4u g0L;
    g0L.x = 1u                      // count = 1 descriptor
          | (1u << 30)              // gather_index_size = 32-bit
          | (1u << 31);             // gather_mode = 1
    g0L.y = ldsAddr;                // lds_addr
    g0L.z = (u32)(gsrc & 0xFFFFFFFFu);               // global_addr[31:0]
    g0L.w = (u32)((gsrc >> 32) & 0x01FFFFFFu)        // global_addr[56:32]
          | (2u << 30);                              // type = 2 ("image")

    v8i g1L;
    g1L[0] = (int)(2u << 16);                // data_size = 2 (4 bytes)
    g1L[1] = (int)((u32)(kH & 0xFFFF) << 16);// tensor_dim0[15:0] = H
    g1L[2] = (int)((u32)(kS & 0xFFFF) << 16);// tensor_dim1[15:0] = S
    g1L[3] = (int)((u32)(kH & 0xFFFF) << 16);// tile_dim0 = H (full row)
    g1L[4] = kRows;                          // tile_dim1 = #valid indices
    g1L[5] = kH;                             // tensor_dim0_stride = H
    g1L[6] = 0;                              // stride hi / dim1_stride (ignored)
    g1L[7] = 0;

    v4i g2L; v4i g3L;                        // 32-bit row indices 0..3 / 4..7
    g2L[0] = idx[0]; g2L[1] = idx[1]; g2L[2] = idx[2]; g2L[3] = idx[3];
    g3L[0] = idx[4]; g3L[1] = idx[5]; g3L[2] = idx[6]; g3L[3] = idx[7];

    v8i zero8 = {0, 0, 0, 0, 0, 0, 0, 0};

    __builtin_amdgcn_tensor_load_to_lds(g0L, g1L, g2L, g3L, zero8, 0);

    // Data must be resident in LDS before the TDM store streams it out.
    __builtin_amdgcn_s_wait_tensorcnt(0);

    // ---------------- Store descriptor: LDS tile -> out[b, p0..p0+7, :] ----
    const u64 gdst = (u64)(uintptr_t)(out + (u64)tileId * (u64)kTileElems);
    v4u g0S;
    g0S.x = 1u;                     // count = 1, normal (non-gather) mode
    g0S.y = ldsAddr;
    g0S.z = (u32)(gdst & 0xFFFFFFFFu);
    g0S.w = (u32)((gdst >> 32) & 0x01FFFFFFu) | (2u << 30);

    v8i g1S;
    g1S[0] = (int)(2u << 16);                        // data_size = 4 bytes
    g1S[1] = (int)((u32)(kTileElems & 0xFFFF) << 16);// tensor_dim0 = 8192
    g1S[2] = (int)(1u << 16);                        // tensor_dim1 = 1
    g1S[3] = (int)((u32)(kTileElems & 0xFFFF) << 16);// tile_dim0 = 8192
    g1S[4] = 1;                                      // tile_dim1 = 1
    g1S[5] = kTileElems;                             // tensor_dim0_stride
    g1S[6] = 0;
    g1S[7] = 0;

    v4i zero4 = {0, 0, 0, 0};
    __builtin_amdgcn_tensor_store_from_lds(g0S, g1S, zero4, zero4, zero8, 0);

    // Drain before wave exit (S_ENDPGM also implies wait-idle).
    __builtin_amdgcn_s_wait_tensorcnt(0);
}

extern "C" void kernel_launch(void* const* d_in, const int* in_sizes, int n_in,
                              void* d_out, int out_size, void* d_ws, size_t ws_size,
                              hipStream_t stream) {
    (void)in_sizes; (void)n_in; (void)d_ws; (void)ws_size; (void)out_size;
    const float* seq = (const float*)d_in[0];          // [B,S,H] f32
    const long long* pos = (const long long*)d_in[1];  // [B,P] i64
    float* out = (float*)d_out;                        // [B,P,H] f32

    const int nTiles = (kB * kP) / kRows;              // 512 workgroups
    OnehotGather_tdm_kernel<<<nTiles, 32, 0, stream>>>(seq, pos, out);
}